// VariationalQuantumCircuit_81372450390375
// MI455X (gfx1250) — compile-verified
//
#include <hip/hip_runtime.h>
#include <math.h>

typedef __attribute__((ext_vector_type(2))) float v2f;
typedef __attribute__((ext_vector_type(8))) float v8f;

#define NQ 12
#define DIM 4096

// D(16x16,f32) = A(16x4,f32) x B(4x16,f32) + C
__device__ __forceinline__ v8f wmma4(v2f a, v2f b, v8f c) {
  return __builtin_amdgcn_wmma_f32_16x16x4_f32(false, a, false, b, (short)0, c,
                                               false, false);
}

// Fast HW trig: v_cos_f32 / v_sin_f32 (angles here are small — well inside
// the fast range; avoids inlining the huge libm Payne-Hanek slow path).
__device__ __forceinline__ void fast_sincos(float a, float* s, float* c) {
  *s = __sinf(a);
  *c = __cosf(a);
}

// RY(theta) = [[c,-s],[s,c]]  (row r, col cb)
__device__ __forceinline__ float ry_elem(int r, int cb, float c, float s) {
  return r ? (cb ? c : s) : (cb ? -s : c);
}

struct LaneCtx {
  int lane;  // 0..31
  int m16;   // lane & 15
  int kh;    // (lane>>4)*2 : lanes 16-31 hold K+2 in A/B f32 fragments
  int rofs;  // (lane>>4)*8 : C/D VGPR v holds rows v / v+8
};

// Apply 16x16 operator M along tensor mode MODE of the 16x16x16 state (in
// place, two independent 16x16x16 block-matmuls per iteration for ILP; the
// pair loop itself stays rolled to keep VGPR pressure < 256).
//  MODE 0: D = M x S[:, blk]        (d0' contraction)
//  MODE 1: D = M x S[blk][k][d2]    (middle mode)
//  MODE 2: D = S_blk x M^T          (d2' contraction; Mf acts as B = M^T)
template <int MODE>
__device__ __forceinline__ void apply_mode(float* cur, const v2f* Mf,
                                           const LaneCtx& L) {
#pragma unroll 1
  for (int bp = 0; bp < 8; ++bp) {
    const int b0 = 2 * bp, b1 = b0 + 1;
    v8f acc0 = {}, acc1 = {};
#pragma unroll
    for (int k0 = 0; k0 < 4; ++k0) {
      const int k = 4 * k0 + L.kh;
      if (MODE == 2) {
        v2f a0, a1;
        a0.x = cur[(b0 * 16 + L.m16) * 16 + k];
        a0.y = cur[(b0 * 16 + L.m16) * 16 + k + 1];
        a1.x = cur[(b1 * 16 + L.m16) * 16 + k];
        a1.y = cur[(b1 * 16 + L.m16) * 16 + k + 1];
        acc0 = wmma4(a0, Mf[k0], acc0);
        acc1 = wmma4(a1, Mf[k0], acc1);
      } else {
        v2f bb0, bb1;
        if (MODE == 0) {
          bb0.x = cur[k * 256 + b0 * 16 + L.m16];
          bb0.y = cur[(k + 1) * 256 + b0 * 16 + L.m16];
          bb1.x = cur[k * 256 + b1 * 16 + L.m16];
          bb1.y = cur[(k + 1) * 256 + b1 * 16 + L.m16];
        } else {
          bb0.x = cur[b0 * 256 + k * 16 + L.m16];
          bb0.y = cur[b0 * 256 + (k + 1) * 16 + L.m16];
          bb1.x = cur[b1 * 256 + k * 16 + L.m16];
          bb1.y = cur[b1 * 256 + (k + 1) * 16 + L.m16];
        }
        acc0 = wmma4(Mf[k0], bb0, acc0);
        acc1 = wmma4(Mf[k0], bb1, acc1);
      }
    }
    __syncthreads();  // all reads of this block pair complete before overwrite
#pragma unroll
    for (int v = 0; v < 8; ++v) {
      if (MODE == 0) {
        cur[(L.rofs + v) * 256 + b0 * 16 + L.m16] = acc0[v];
        cur[(L.rofs + v) * 256 + b1 * 16 + L.m16] = acc1[v];
      } else if (MODE == 1) {
        cur[b0 * 256 + (L.rofs + v) * 16 + L.m16] = acc0[v];
        cur[b1 * 256 + (L.rofs + v) * 16 + L.m16] = acc1[v];
      } else {
        cur[(b0 * 16 + L.rofs + v) * 16 + L.m16] = acc0[v];
        cur[(b1 * 16 + L.rofs + v) * 16 + L.m16] = acc1[v];
      }
    }
  }
  __syncthreads();  // mode fully written before next consumer
}

// One qubit-group of one layer: build theta operator, combine with the
// per-batch input operator (M = Uth @ Uin, 4 chained WMMAs), apply along mode G.
template <int G>
__device__ __forceinline__ void do_group(const float* __restrict__ thetas,
                                         int l, float* cur, float* sUth,
                                         float* sM, const float (*sUin)[256],
                                         const LaneCtx& L) {
  float ct[4], st[4];
#pragma unroll
  for (int u = 0; u < 4; ++u) {
    float a = 0.5f * thetas[l * NQ + 4 * G + u];
    fast_sincos(a, &st[u], &ct[u]);
  }
#pragma unroll
  for (int t = 0; t < 8; ++t) {
    int e = L.lane * 8 + t, i = e >> 4, j = e & 15;
    float v = 1.0f;
#pragma unroll
    for (int u = 0; u < 4; ++u)
      v *= ry_elem((i >> (3 - u)) & 1, (j >> (3 - u)) & 1, ct[u], st[u]);
    sUth[e] = v;
  }
  __syncthreads();

  v8f acc = {};
#pragma unroll
  for (int k0 = 0; k0 < 4; ++k0) {
    const int k = 4 * k0 + L.kh;
    v2f a, bb;
    a.x = sUth[L.m16 * 16 + k];
    a.y = sUth[L.m16 * 16 + k + 1];
    bb.x = sUin[G][k * 16 + L.m16];
    bb.y = sUin[G][(k + 1) * 16 + L.m16];
    acc = wmma4(a, bb, acc);
  }
#pragma unroll
  for (int v = 0; v < 8; ++v) sM[(L.rofs + v) * 16 + L.m16] = acc[v];
  __syncthreads();

  // M fragments: A for modes 0/1 (D = M x S), and B = M^T for mode 2
  v2f Mf[4];
#pragma unroll
  for (int k0 = 0; k0 < 4; ++k0) {
    Mf[k0].x = sM[L.m16 * 16 + 4 * k0 + L.kh];
    Mf[k0].y = sM[L.m16 * 16 + 4 * k0 + L.kh + 1];
  }
  apply_mode<G>(cur, Mf, L);
}

// 32 threads/block (one wave); min 4 waves/SIMD => VGPR cap 256, avoiding the
// S_SET_VGPR_MSB high-VGPR addressing path.
__global__ __launch_bounds__(32, 4) void vqc_kernel(
    const float* __restrict__ angles,  // (B, 12)
    const float* __restrict__ thetas,  // (L, 12)
    float* __restrict__ out,           // (B, 12)
    int L) {
  const int b = blockIdx.x;
  LaneCtx lc;
  lc.lane = threadIdx.x;
  lc.m16 = lc.lane & 15;
  lc.kh = (lc.lane >> 4) << 1;
  lc.rofs = (lc.lane >> 4) << 3;

  __shared__ float sA[DIM];       // state ping
  __shared__ float sB[DIM];       // state pong (CNOT gather target)
  __shared__ float sUin[3][256];  // per-batch input-angle group operators
  __shared__ float sUth[256];     // theta group operator (rebuilt per (l,g))
  __shared__ float sM[256];       // combined operator

  // ---- per-batch cos/sin of half-angles (HW trig) ----
  float ca[NQ], sa[NQ];
#pragma unroll
  for (int q = 0; q < NQ; ++q) {
    float a = 0.5f * angles[b * NQ + q];
    fast_sincos(a, &sa[q], &ca[q]);
  }

  // ---- Uin[g] = RY(a[4g]) (x) ... (x) RY(a[4g+3]) ----
#pragma unroll
  for (int g = 0; g < 3; ++g)
#pragma unroll
    for (int t = 0; t < 8; ++t) {
      int e = lc.lane * 8 + t, i = e >> 4, j = e & 15;
      float v = 1.0f;
#pragma unroll
      for (int u = 0; u < 4; ++u) {
        int q = 4 * g + u;
        v *= ry_elem((i >> (3 - u)) & 1, (j >> (3 - u)) & 1, ca[q], sa[q]);
      }
      sUin[g][e] = v;
    }

  // ---- init |0...0> ----
#pragma unroll 1
  for (int e = lc.lane; e < DIM; e += 32) sA[e] = 0.0f;
  __syncthreads();
  if (lc.lane == 0) sA[0] = 1.0f;
  __syncthreads();

  float* cur = sA;
  float* nxt = sB;

#pragma unroll 1
  for (int l = 0; l < L; ++l) {
    do_group<0>(thetas, l, cur, sUth, sM, sUin, lc);
    do_group<1>(thetas, l, cur, sUth, sM, sUin, lc);
    do_group<2>(thetas, l, cur, sUth, sM, sUin, lc);

    // ---- all 11 CNOTs composed into one gather, ping-pong ----
#pragma unroll 1
    for (int e = lc.lane; e < DIM; e += 32) {
      int j = e;
#pragma unroll
      for (int q = NQ - 2; q >= 0; --q)  // innermost (q=10) applied first
        j ^= ((j >> (NQ - 1 - q)) & 1) << (NQ - 2 - q);
      nxt[e] = cur[j];
    }
    __syncthreads();
    float* tmp = cur;
    cur = nxt;
    nxt = tmp;
  }

  // ---- out[b][q] = sum_i state[i]^2 * (1 - 2*bit_q(i)) ----
  float acc[NQ];
#pragma unroll
  for (int q = 0; q < NQ; ++q) acc[q] = 0.0f;
#pragma unroll 1
  for (int e = lc.lane; e < DIM; e += 32) {
    float s = cur[e];
    float p = s * s;
#pragma unroll
    for (int q = 0; q < NQ; ++q)
      acc[q] += ((e >> (NQ - 1 - q)) & 1) ? -p : p;
  }
#pragma unroll
  for (int q = 0; q < NQ; ++q) {
    float v = acc[q];
    for (int off = 16; off > 0; off >>= 1) v += __shfl_xor(v, off, 32);
    if (lc.lane == 0) out[b * NQ + q] = v;
  }
}

extern "C" void kernel_launch(void* const* d_in, const int* in_sizes, int n_in,
                              void* d_out, int out_size, void* d_ws,
                              size_t ws_size, hipStream_t stream) {
  (void)n_in; (void)out_size; (void)d_ws; (void)ws_size;
  const float* angles = (const float*)d_in[0];  // (B, 12) f32
  const float* thetas = (const float*)d_in[1];  // (L, 12) f32
  float* out = (float*)d_out;                   // (B, 12) f32
  const int B = in_sizes[0] / NQ;
  const int L = in_sizes[1] / NQ;
  vqc_kernel<<<dim3(B), dim3(32), 0, stream>>>(angles, thetas, out, L);
}